// ReKT_21320217657962
// MI455X (gfx1250) — compile-verified
//
#include <hip/hip_runtime.h>
#include <cmath>

#define B_ 256
#define L_ 200
#define Q_ 20000
#define C_ 500
#define D_ 256
#define S_ 199

typedef float v2f __attribute__((ext_vector_type(2)));
typedef float v8f __attribute__((ext_vector_type(8)));

__device__ __forceinline__ v2f ld2(const float* p) { return *(const v2f*)p; }
__device__ __forceinline__ float sigmoidf_(float x) { return 1.0f / (1.0f + expf(-x)); }

// ---------------------------------------------------------------------------
// Kernel A: per-step gather / setup.  One block per batch row b, 256 threads.
// Builds gate inputs [state, time_gate] (B x 512) for p/c/a, the "final"
// tail (pe) and x = pe + ans_embed, and bumps lpt/lct after the gather.
// ---------------------------------------------------------------------------
__global__ void rekt_setup(int t,
    const int* __restrict__ qseq, const int* __restrict__ cseq, const int* __restrict__ aseq,
    const float* __restrict__ pro_embed, const float* __restrict__ skill_embed,
    const float* __restrict__ ans_embed, const float* __restrict__ time_embed,
    const float* __restrict__ akt_pro_diff, const float* __restrict__ akt_pro_change,
    int* lpt, int* lct,
    const float* __restrict__ ps, const float* __restrict__ ss, const float* __restrict__ ast,
    float* Ap, float* Ac, float* Aa, float* Fin, float* X)
{
    int b = blockIdx.x;
    int d = threadIdx.x;                       // 0..255
    int ip = qseq[b * L_ + t + 1];
    int ic = cseq[b * L_ + t + 1];
    int ia = aseq[b * L_ + t + 1];
    int lp_t = lpt[b * Q_ + ip];
    int lc_t = lct[b * C_ + ic];

    float lp_s = ps[((size_t)b * S_ + lp_t) * D_ + d];
    float lc_s = ss[((size_t)b * S_ + lc_t) * D_ + d];

    Ap[b * 512 + d]       = lp_s;
    Ap[b * 512 + 256 + d] = time_embed[(t - lp_t) * D_ + d];
    Ac[b * 512 + d]       = lc_s;
    Ac[b * 512 + 256 + d] = time_embed[(t - lc_t) * D_ + d];
    Aa[b * 512 + d]       = ast[b * D_ + d];
    Aa[b * 512 + 256 + d] = time_embed[1 * D_ + d];          // atge = time_embed[1]

    float pe = pro_embed[(size_t)ip * D_ + d] + skill_embed[ic * D_ + d]
             + akt_pro_diff[ip] * akt_pro_change[ic * D_ + d];
    Fin[b * 1024 + 768 + d] = pe;                            // final[:, 768:1024] = pe
    X[b * 256 + d] = pe + ans_embed[ia * D_ + d];

    __syncthreads();
    if (d == 0) { lpt[b * Q_ + ip] = t; lct[b * C_ + ic] = t; }
}

// ---------------------------------------------------------------------------
// Kernel B: the three gate GEMMs (M=256, N=256, K=512), f32 WMMA 16x16x4.
// grid = (16, 3): x = row tile (16 rows), y = which gate (0:p 1:c 2:a).
// block = 512 threads = 16 waves; wave w computes column tile n0 = 16*w.
// Epilogue: S = state * sigmoid(Y + bias); also written into Fin slot.
// ---------------------------------------------------------------------------
__global__ void rekt_gates(const float* __restrict__ Abase,
    const float* __restrict__ Wpf, const float* __restrict__ bpf,
    const float* __restrict__ Wsf, const float* __restrict__ bsf,
    const float* __restrict__ Waf, const float* __restrict__ baf,
    float* Sbase, float* Fin)
{
    int y = blockIdx.y;                                      // 0:p 1:c 2:a
    const float* Ain = Abase + (size_t)y * B_ * 512;
    const float* W   = (y == 0) ? Wpf : (y == 1) ? Wsf : Waf;
    const float* bi  = (y == 0) ? bpf : (y == 1) ? bsf : baf;
    int slot = (y + 1) % 3;                                  // Sbase/Fin order: a,p,c
    float* Sout = Sbase + (size_t)slot * B_ * 256;

    int m0   = blockIdx.x * 16;
    int wv   = threadIdx.x >> 5;
    int lane = threadIdx.x & 31;
    int h = lane >> 4, j = lane & 15;
    int n0 = wv * 16;

    const float* arow = Ain + (m0 + j) * 512;
    const float* wrow = W   + (n0 + j) * 512;

    v8f acc = {};
    for (int k0 = 0; k0 < 512; k0 += 4) {
        v2f a = ld2(arow + k0 + 2 * h);
        v2f b = ld2(wrow + k0 + 2 * h);
        acc = __builtin_amdgcn_wmma_f32_16x16x4_f32(false, a, false, b, (short)0, acc, false, false);
    }
    for (int r = 0; r < 8; ++r) {
        int m = m0 + r + 8 * h;
        int n = n0 + j;
        float sv = Ain[m * 512 + n] * sigmoidf_(acc[r] + bi[n]);
        Sout[m * 256 + n] = sv;
        Fin[m * 1024 + slot * 256 + n] = sv;
    }
}

// ---------------------------------------------------------------------------
// Kernel C: grid = (16, 4).
//   y == 0 : Wo1 GEMM (K=1024) + ReLU -> LDS, fused Wo2 dot + sigmoid -> out
//   y == 1 : a_new = Sa + tanh([Sa,x] @ Was^T + bas) -> ast
//   y == 2 : p_new = Sp + tanh([Sp,x] @ Wps^T + bps) -> ps[:, t]
//   y == 3 : c_new = Sc + tanh([Sc,x] @ Wss^T + bss) -> ss[:, t]
// ---------------------------------------------------------------------------
__global__ void rekt_update_out(int t,
    const float* __restrict__ Sbase, const float* __restrict__ Fin, const float* __restrict__ X,
    const float* __restrict__ Wo1, const float* __restrict__ bo1,
    const float* __restrict__ Wo2, const float* __restrict__ bo2,
    const float* __restrict__ Was, const float* __restrict__ bas,
    const float* __restrict__ Wps, const float* __restrict__ bps,
    const float* __restrict__ Wss, const float* __restrict__ bss,
    float* ast, float* ps, float* ss, float* out)
{
    __shared__ float Hs[16 * 256];
    int y    = blockIdx.y;
    int m0   = blockIdx.x * 16;
    int wv   = threadIdx.x >> 5;
    int lane = threadIdx.x & 31;
    int h = lane >> 4, j = lane & 15;
    int n0 = wv * 16;

    if (y == 0) {
        const float* arow = Fin + (m0 + j) * 1024;
        const float* wrow = Wo1 + (n0 + j) * 1024;
        v8f acc = {};
        for (int k0 = 0; k0 < 1024; k0 += 4) {
            v2f a = ld2(arow + k0 + 2 * h);
            v2f b = ld2(wrow + k0 + 2 * h);
            acc = __builtin_amdgcn_wmma_f32_16x16x4_f32(false, a, false, b, (short)0, acc, false, false);
        }
        for (int r = 0; r < 8; ++r) {
            int rr = r + 8 * h;
            int n  = n0 + j;
            float hv = acc[r] + bo1[n];
            Hs[rr * 256 + n] = hv > 0.0f ? hv : 0.0f;
        }
        __syncthreads();
        // Fused Wo2: wave wv reduces row wv (256 cols) against Wo2.
        int row = wv;
        float part = 0.0f;
        for (int i = 0; i < 8; ++i) {
            int col = lane + 32 * i;
            part += Hs[row * 256 + col] * Wo2[col];
        }
        for (int off = 16; off > 0; off >>= 1)
            part += __shfl_xor(part, off, 32);
        if (lane == 0)
            out[(m0 + row) * S_ + t] = sigmoidf_(part + bo2[0]);
    } else {
        const float* Sstate = Sbase + (size_t)(y - 1) * B_ * 256;  // y=1:Sa y=2:Sp y=3:Sc
        const float* W  = (y == 1) ? Was : (y == 2) ? Wps : Wss;
        const float* bi = (y == 1) ? bas : (y == 2) ? bps : bss;
        const float* s1   = Sstate + (m0 + j) * 256;
        const float* s2   = X      + (m0 + j) * 256;
        const float* wrow = W + (n0 + j) * 512;
        v8f acc = {};
        for (int k0 = 0; k0 < 512; k0 += 4) {
            const float* ap = (k0 < 256) ? (s1 + k0) : (s2 + (k0 - 256));
            v2f a = ld2(ap + 2 * h);
            v2f b = ld2(wrow + k0 + 2 * h);
            acc = __builtin_amdgcn_wmma_f32_16x16x4_f32(false, a, false, b, (short)0, acc, false, false);
        }
        for (int r = 0; r < 8; ++r) {
            int m = m0 + r + 8 * h;
            int n = n0 + j;
            float nv = Sstate[m * 256 + n] + tanhf(acc[r] + bi[n]);
            if (y == 1)      ast[m * 256 + n] = nv;
            else if (y == 2) ps[((size_t)m * S_ + t) * 256 + n] = nv;
            else             ss[((size_t)m * S_ + t) * 256 + n] = nv;
        }
    }
}

// ---------------------------------------------------------------------------
// Init kernels (run every launch; kernel_launch must be deterministic).
// ---------------------------------------------------------------------------
__global__ void rekt_init_idx(int* lpt, int* lct)
{
    long n = (long)B_ * Q_ + (long)B_ * C_;
    for (long i = blockIdx.x * (long)blockDim.x + threadIdx.x; i < n;
         i += (long)gridDim.x * blockDim.x) {
        if (i < (long)B_ * Q_) lpt[i] = 0;
        else                   lct[i - (long)B_ * Q_] = 0;
    }
}

__global__ void rekt_init_states(const float* __restrict__ pro_state0,
                                 const float* __restrict__ skill_state0,
                                 const float* __restrict__ ls_state,
                                 float* ps, float* ss, float* ast)
{
    long n = (long)B_ * S_ * D_;
    for (long i = blockIdx.x * (long)blockDim.x + threadIdx.x; i < n;
         i += (long)gridDim.x * blockDim.x) {
        long sd = i % ((long)S_ * D_);
        ps[i] = pro_state0[sd];
        ss[i] = skill_state0[sd];
        if (i < (long)B_ * D_) ast[i] = ls_state[i % D_];
    }
}

// ---------------------------------------------------------------------------
extern "C" void kernel_launch(void* const* d_in, const int* in_sizes, int n_in,
                              void* d_out, int out_size, void* d_ws, size_t ws_size,
                              hipStream_t stream)
{
    const int*   qseq           = (const int*)  d_in[0];
    const int*   cseq           = (const int*)  d_in[1];
    const int*   aseq           = (const int*)  d_in[2];
    const float* pro_embed      = (const float*)d_in[3];
    const float* skill_embed    = (const float*)d_in[4];
    const float* ans_embed      = (const float*)d_in[5];
    const float* time_embed     = (const float*)d_in[6];
    const float* ls_state       = (const float*)d_in[7];
    const float* pro_state0     = (const float*)d_in[8];
    const float* skill_state0   = (const float*)d_in[9];
    const float* akt_pro_diff   = (const float*)d_in[10];
    const float* akt_pro_change = (const float*)d_in[11];
    const float* Wpf = (const float*)d_in[12]; const float* bpf = (const float*)d_in[13];
    const float* Wps = (const float*)d_in[14]; const float* bps = (const float*)d_in[15];
    const float* Wsf = (const float*)d_in[16]; const float* bsf = (const float*)d_in[17];
    const float* Wss = (const float*)d_in[18]; const float* bss = (const float*)d_in[19];
    const float* Waf = (const float*)d_in[20]; const float* baf = (const float*)d_in[21];
    const float* Was = (const float*)d_in[22]; const float* bas = (const float*)d_in[23];
    const float* Wo1 = (const float*)d_in[24]; const float* bo1 = (const float*)d_in[25];
    const float* Wo2 = (const float*)d_in[26]; const float* bo2 = (const float*)d_in[27];

    float* out = (float*)d_out;                       // (B, S)

    // Workspace layout (all offsets 8-byte aligned).
    char* ws = (char*)d_ws;
    size_t off = 0;
    int*   lpt   = (int*)(ws + off);  off += (size_t)B_ * Q_ * 4;        // 20,480,000
    int*   lct   = (int*)(ws + off);  off += (size_t)B_ * C_ * 4;        //    512,000
    float* ps    = (float*)(ws + off); off += (size_t)B_ * S_ * D_ * 4;  // 52,166,656
    float* ss    = (float*)(ws + off); off += (size_t)B_ * S_ * D_ * 4;  // 52,166,656
    float* ast   = (float*)(ws + off); off += (size_t)B_ * D_ * 4;
    float* Abase = (float*)(ws + off); off += 3 * (size_t)B_ * 512 * 4;  // Ap, Ac, Aa
    float* Sbase = (float*)(ws + off); off += 3 * (size_t)B_ * 256 * 4;  // Sa, Sp, Sc
    float* Fin   = (float*)(ws + off); off += (size_t)B_ * 1024 * 4;
    float* X     = (float*)(ws + off); off += (size_t)B_ * 256 * 4;
    (void)ws_size; (void)in_sizes; (void)n_in; (void)out_size;

    float* Ap = Abase;
    float* Ac = Abase + (size_t)B_ * 512;
    float* Aa = Abase + 2 * (size_t)B_ * 512;

    rekt_init_idx<<<1024, 256, 0, stream>>>(lpt, lct);
    rekt_init_states<<<2048, 256, 0, stream>>>(pro_state0, skill_state0, ls_state, ps, ss, ast);

    for (int t = 0; t < S_; ++t) {
        rekt_setup<<<B_, 256, 0, stream>>>(t, qseq, cseq, aseq,
            pro_embed, skill_embed, ans_embed, time_embed, akt_pro_diff, akt_pro_change,
            lpt, lct, ps, ss, ast, Ap, Ac, Aa, Fin, X);
        rekt_gates<<<dim3(16, 3), 512, 0, stream>>>(Abase,
            Wpf, bpf, Wsf, bsf, Waf, baf, Sbase, Fin);
        rekt_update_out<<<dim3(16, 4), 512, 0, stream>>>(t,
            Sbase, Fin, X, Wo1, bo1, Wo2, bo2,
            Was, bas, Wps, bps, Wss, bss,
            ast, ps, ss, out);
    }
}